// WFAConstrainedLoss_52828097741444
// MI455X (gfx1250) — compile-verified
//
#include <hip/hip_runtime.h>
#include <hip/hip_bf16.h>
#include <math.h>

// ---- problem constants (match reference) ----
#define NWL        112      // raw wavelength count
#define NW         111      // sliced length [0:111]
#define NATM       84       // N_LOGTAU * N_MAGS
#define BLOS_TH    100.0f
#define BASE_TH    0.0004f
#define EPSF       1e-10f
// WFA_C * LAMBDA0^2 * G_FACTOR
#define WFA_DENOM  (4.6686e-13f * 6301.5f * 6301.5f * 1.5f)

#define THREADS        128      // 4 wave32 waves
#define WAVES          4
#define PIX_PER_BLOCK  64       // 16 pixels per wave
// LDS row strides: stride%64 coprime with 64 -> conflict-free 16-row gathers
#define SI_STRIDE      115      // holds J[-1..112] + pad  (114 used), 115%64=51
#define SV_STRIDE      113      // holds V[0..111]         (112 used), 113%64=49

typedef float v2f __attribute__((ext_vector_type(2)));
typedef float v8f __attribute__((ext_vector_type(8)));

__global__ __launch_bounds__(THREADS)
void wfa_main_kernel(const float* __restrict__ stokes,   // (B,4,112)
                     const float* __restrict__ pred,     // (B,84)
                     const float* __restrict__ targ,     // (B,84)
                     const float* __restrict__ wl,       // (112,)
                     float* __restrict__ acc,            // [0]=abs_sum [1]=wfa_sum [2]=cnt
                     int B)
{
    // sI row p holds J[-1..112] at [0..113]:  J[w] = I[w] at index w+1, halos at 0/112, pad at 113
    __shared__ float sI[PIX_PER_BLOCK * SI_STRIDE];
    __shared__ float sV[PIX_PER_BLOCK * SV_STRIDE];
    __shared__ float srd[NWL];          // 0.5/dwl[k] for k<111, 0 at k=111
    __shared__ float sums[WAVES][4][16];
    __shared__ float sred[3];           // abs_sum, wfa_sum, cnt (block partials)

    const int tid      = threadIdx.x;
    const int base_pix = blockIdx.x * PIX_PER_BLOCK;

    if (tid < 3) sred[tid] = 0.0f;

    // rd[k] = 0.5 / gradient(wl)[k]; edge-consistent with the J-halo trick below.
    if (tid < NWL) {
        float r = 0.0f;
        if (tid < NW) {
            float g = (tid == 0)      ? (wl[1] - wl[0])
                    : (tid == NW - 1) ? (wl[NW - 1] - wl[NW - 2])
                    :                   0.5f * (wl[tid + 1] - wl[tid - 1]);
            r = 0.5f / g;
        }
        srd[tid] = r;                   // srd[111] = 0 kills the k=111 pad column
    }

    // Stage Stokes I (comp 0) and V (comp 3) rows into LDS, coalesced.
    for (int idx = tid; idx < PIX_PER_BLOCK * NWL; idx += THREADS) {
        int p = idx / NWL, w = idx % NWL;
        int gp = base_pix + p;
        float iv = 0.0f, vv = 0.0f;
        if (gp < B && w < NW) {
            size_t rb = (size_t)gp * (4 * NWL);
            iv = stokes[rb + w];                 // I
            vv = stokes[rb + 3 * NWL + w];       // V
        }
        if (w < NW) sI[p * SI_STRIDE + 1 + w] = iv;
        sV[p * SV_STRIDE + w] = vv;              // V[111] = 0 (pad column)
    }
    __syncthreads();

    // Halos: J[-1] = 2*I[0]-I[1], J[111] = 2*I[110]-I[109], finite pad at J[112].
    // Then (J[k+1]-J[k-1]) * rd[k] == np.gradient(I)[k]/dwl[k] uniformly for k=0..110.
    if (tid < PIX_PER_BLOCK) {
        float* r = &sI[tid * SI_STRIDE];
        r[0]   = 2.0f * r[1]   - r[2];
        r[112] = 2.0f * r[111] - r[110];
        r[113] = 0.0f;                           // must be finite: NaN*0 = NaN
    }
    __syncthreads();

    // ---- per-wave WMMA reduction: 16 pixels (M), 4 wavelengths per step (K) ----
    const int wv   = tid >> 5;
    const int lane = tid & 31;
    const int m    = lane & 15;          // A-matrix row owned by this lane
    const int kb   = (lane >> 4) * 2;    // lanes 16-31 hold K=2,3 of each chunk
    const float* Ib = &sI[(wv * 16 + m) * SI_STRIDE] + kb;   // Ib[k0] = J[k0+kb-1]
    const float* Vb = &sV[(wv * 16 + m) * SV_STRIDE] + kb;
    const float* rb = srd + kb;

    v8f accD = {}, accV = {}, accDD = {}, accDV = {};
    const v2f ones = {1.0f, 1.0f};

    #pragma unroll
    for (int k0 = 0; k0 < NWL; k0 += 4) {
        // this lane's two K-slots are k = k0+kb, k0+kb+1
        float j0 = Ib[k0], j1 = Ib[k0 + 1], j2 = Ib[k0 + 2], j3 = Ib[k0 + 3];
        float r0 = rb[k0], r1 = rb[k0 + 1];
        float v0 = Vb[k0], v1 = Vb[k0 + 1];
        float d0 = (j2 - j0) * r0;
        float d1 = (j3 - j1) * r1;
        v2f aD  = {d0,      d1};
        v2f aV  = {v0,      v1};
        v2f aDD = {d0 * d0, d1 * d1};
        v2f aDV = {d0 * v0, d1 * v1};
        // D = A x ones + C : running row-sums for 16 pixels, fp32 accumulate
        accD  = __builtin_amdgcn_wmma_f32_16x16x4_f32(false, aD,  false, ones, (short)0, accD,  false, false);
        accV  = __builtin_amdgcn_wmma_f32_16x16x4_f32(false, aV,  false, ones, (short)0, accV,  false, false);
        accDD = __builtin_amdgcn_wmma_f32_16x16x4_f32(false, aDD, false, ones, (short)0, accDD, false, false);
        accDV = __builtin_amdgcn_wmma_f32_16x16x4_f32(false, aDV, false, ones, (short)0, accDV, false, false);
    }

    // C/D layout: lane 0 VGPR r -> D[r][0] (rows 0-7), lane 16 VGPR r -> D[8+r][0]
    if (lane == 0 || lane == 16) {
        int ro = (lane == 0) ? 0 : 8;
        #pragma unroll
        for (int r = 0; r < 8; ++r) {
            sums[wv][0][ro + r] = accD[r];
            sums[wv][1][ro + r] = accV[r];
            sums[wv][2][ro + r] = accDD[r];
            sums[wv][3][ro + r] = accDV[r];
        }
    }
    __syncthreads();

    // ---- per-pixel WFA regression + masked log-ratio ----
    if (lane < 16) {
        int gp = base_pix + wv * 16 + lane;
        if (gp < B) {
            float sd  = sums[wv][0][lane];
            float sv  = sums[wv][1][lane];
            float sdd = sums[wv][2][lane];
            float sdv = sums[wv][3][lane];
            float n   = (float)NW;
            float p0  = (n * sdv - sd * sv) / (n * sdd - sd * sd);
            float wfa = -p0 / WFA_DENOM;
            if (fabsf(wfa) < BLOS_TH) {
                size_t ab = (size_t)gp * NATM;
                // atm[:, tau={0,1,2}, -1] -> flat indices 3,7,11
                float pb = (pred[ab + 3] + pred[ab + 7] + pred[ab + 11]) * (1.0f / 3.0f);
                float diff = fabsf(log10f(fabsf(pb) + EPSF) - log10f(fabsf(wfa) + EPSF));
                atomicAdd(&sred[1], diff);
                atomicAdd(&sred[2], 1.0f);
            }
        }
    }

    // ---- base-loss partial: sum |pred - targ| over this block's pixels ----
    {
        int npix = B - base_pix;
        if (npix > PIX_PER_BLOCK) npix = PIX_PER_BLOCK;
        int tot = npix * NATM;
        size_t gbase = (size_t)base_pix * NATM;
        float a = 0.0f;
        for (int idx = tid; idx < tot; idx += THREADS)
            a += fabsf(pred[gbase + idx] - targ[gbase + idx]);
        atomicAdd(&sred[0], a);
    }
    __syncthreads();

    if (tid == 0) {
        atomicAdd(&acc[0], sred[0]);
        atomicAdd(&acc[1], sred[1]);
        atomicAdd(&acc[2], sred[2]);
    }
}

__global__ void wfa_finalize_kernel(const float* __restrict__ acc,
                                    float* __restrict__ out,
                                    float inv_total)
{
    float base  = acc[0] * inv_total;
    float cnt   = acc[2];
    float wfa_m = acc[1] / fmaxf(cnt, 1.0f);
    bool apply  = (base < BASE_TH) && (cnt > 0.0f);
    out[0] = apply ? (0.5f * base + 0.5f * wfa_m) : base;   // total_loss
    out[1] = base;                                          // base_loss
    out[2] = apply ? wfa_m : 0.0f;                          // wfa_loss
}

extern "C" void kernel_launch(void* const* d_in, const int* in_sizes, int n_in,
                              void* d_out, int out_size, void* d_ws, size_t ws_size,
                              hipStream_t stream)
{
    const float* stokes = (const float*)d_in[0];   // (B,4,112) fp32
    const float* pred   = (const float*)d_in[1];   // (B,84)    fp32
    const float* targ   = (const float*)d_in[2];   // (B,84)    fp32
    const float* wl     = (const float*)d_in[3];   // (112,)    fp32

    int B = in_sizes[0] / (4 * NWL);
    float* acc = (float*)d_ws;

    hipMemsetAsync(acc, 0, 3 * sizeof(float), stream);

    int blocks = (B + PIX_PER_BLOCK - 1) / PIX_PER_BLOCK;
    wfa_main_kernel<<<blocks, THREADS, 0, stream>>>(stokes, pred, targ, wl, acc, B);

    float inv_total = 1.0f / ((float)B * (float)NATM);
    wfa_finalize_kernel<<<1, 1, 0, stream>>>(acc, (float*)d_out, inv_total);
}